// ComplexTransformerEncoder_5248450035810
// MI455X (gfx1250) — compile-verified
//
#include <hip/hip_runtime.h>
#include <math.h>

// ---------------- problem constants (match reference) ----------------
#define BB 4
#define NN 1024
#define FF 512
#define EE 512
#define HH 8
#define CC 64
#define BN (BB * NN)                 // 4096
#define INV_DIV 0.08838834764831845f // 1/sqrt(2*C) = 1/sqrt(128)

typedef __attribute__((ext_vector_type(2))) float v2f;
typedef __attribute__((ext_vector_type(4))) float v4f;
typedef __attribute__((ext_vector_type(8))) float v8f;

static __device__ __forceinline__ v8f wmma_f32(v2f a, v2f b, v8f c) {
  // V_WMMA_F32_16X16X4_F32 : D = A(16x4) * B(4x16) + C(16x16)
  return __builtin_amdgcn_wmma_f32_16x16x4_f32(
      /*neg_a=*/false, a, /*neg_b=*/false, b,
      /*c_mod=*/(short)0, c, /*reuse_a=*/false, /*reuse_b=*/false);
}
static __device__ __forceinline__ v2f vlo(v4f x) { return __builtin_shufflevector(x, x, 0, 1); }
static __device__ __forceinline__ v2f vhi(v4f x) { return __builtin_shufflevector(x, x, 2, 3); }

// =====================================================================
// Expand complex weight W[e][f] = (wr,wi) into Wx[e][f] = {wr,-wi,wi,wr}
// so one b128 load supplies both B_real and B_imag WMMA fragments.
// =====================================================================
__global__ __launch_bounds__(256) void expand_weight(
    const float* __restrict__ W, float* __restrict__ Wx, int total) {
  const int i = blockIdx.x * 256 + threadIdx.x;
  if (i < total) {
    const float2 w = ((const float2*)W)[i];
    ((float4*)Wx)[i] = make_float4(w.x, -w.y, w.y, w.x);
  }
}

// =====================================================================
// Complex NT-GEMM with bias:  Y[m][e] = sum_f X[m][f] * W[e][f] + bias[e]
// Interleaved-K trick: A fragment = raw contiguous float2 of X.
// B comes pre-expanded ({wr,-wi,wi,wr} per element) -> one v4f load per
// k-step per tile, fragments are subregisters (no construction VALU).
// One wave computes a 16x32 output slab; software-pipelined k-loop.
// expandScore!=0: write Y in score-expanded form {re,im,-im,re} (for K).
// =====================================================================
__global__ __launch_bounds__(32) void cgemm_nt_bias(
    const float* __restrict__ X, const float* __restrict__ Wx,
    const float* __restrict__ bias, float* __restrict__ Y,
    int Nout, int K /* complex */, int expandScore) {
  const int ntn  = Nout >> 5;
  const int tm   = blockIdx.x / ntn;
  const int tn2  = blockIdx.x % ntn;
  const int lane = threadIdx.x;
  const int half = lane >> 4;
  const int l16  = lane & 15;

  const int arow = tm * 16 + l16;
  const int e0   = tn2 * 32 + l16;
  const int e1   = e0 + 16;

  const float* __restrict__ pa  = X + (size_t)arow * (2 * K) + 2 * half;
  const float* __restrict__ pb0 = Wx + ((size_t)e0 * K + half) * 4;
  const float* __restrict__ pb1 = Wx + ((size_t)e1 * K + half) * 4;

  v8f aR0 = {}, aI0 = {}, aR1 = {}, aI1 = {};
  v2f aC[4]; v4f b0C[4], b1C[4];
#pragma unroll
  for (int u = 0; u < 4; ++u) {
    aC[u]  = *(const v2f*)(pa + 4 * u);
    b0C[u] = *(const v4f*)(pb0 + 8 * u);
    b1C[u] = *(const v4f*)(pb1 + 8 * u);
  }
  const int iters = K / 8;        // 8 complex k per iteration (4 wmma steps)
  for (int it = 0; it < iters - 1; ++it) {
    pa += 16; pb0 += 32; pb1 += 32;
    v2f aN[4]; v4f b0N[4], b1N[4];
#pragma unroll
    for (int u = 0; u < 4; ++u) {
      aN[u]  = *(const v2f*)(pa + 4 * u);
      b0N[u] = *(const v4f*)(pb0 + 8 * u);
      b1N[u] = *(const v4f*)(pb1 + 8 * u);
    }
#pragma unroll
    for (int u = 0; u < 4; ++u) {
      aR0 = wmma_f32(aC[u], vlo(b0C[u]), aR0);
      aI0 = wmma_f32(aC[u], vhi(b0C[u]), aI0);
      aR1 = wmma_f32(aC[u], vlo(b1C[u]), aR1);
      aI1 = wmma_f32(aC[u], vhi(b1C[u]), aI1);
    }
#pragma unroll
    for (int u = 0; u < 4; ++u) { aC[u] = aN[u]; b0C[u] = b0N[u]; b1C[u] = b1N[u]; }
  }
#pragma unroll
  for (int u = 0; u < 4; ++u) {
    aR0 = wmma_f32(aC[u], vlo(b0C[u]), aR0);
    aI0 = wmma_f32(aC[u], vhi(b0C[u]), aI0);
    aR1 = wmma_f32(aC[u], vlo(b1C[u]), aR1);
    aI1 = wmma_f32(aC[u], vhi(b1C[u]), aI1);
  }

  const float2 bv0 = *(const float2*)(bias + 2 * e0);
  const float2 bv1 = *(const float2*)(bias + 2 * e1);
  if (expandScore == 0) {
#pragma unroll
    for (int r = 0; r < 8; ++r) {
      const int m = tm * 16 + r + 8 * half;  // D layout: M = vgpr + 8*(L/16)
      float2* y0 = (float2*)(Y + 2 * ((size_t)m * Nout + e0));
      float2* y1 = (float2*)(Y + 2 * ((size_t)m * Nout + e1));
      *y0 = make_float2(aR0[r] + bv0.x, aI0[r] + bv0.y);
      *y1 = make_float2(aR1[r] + bv1.x, aI1[r] + bv1.y);
    }
  } else {
#pragma unroll
    for (int r = 0; r < 8; ++r) {
      const int m = tm * 16 + r + 8 * half;
      const float re0 = aR0[r] + bv0.x, im0 = aI0[r] + bv0.y;
      const float re1 = aR1[r] + bv1.x, im1 = aI1[r] + bv1.y;
      float4* y0 = (float4*)(Y + 4 * ((size_t)m * Nout + e0));
      float4* y1 = (float4*)(Y + 4 * ((size_t)m * Nout + e1));
      *y0 = make_float4(re0, im0, -im0, re0);   // {br, bi, -bi, br}
      *y1 = make_float4(re1, im1, -im1, re1);
    }
  }
}

// =====================================================================
// Attention scores: S[n][m] = sum_c q[n][c] * conj(k[m][c]) / sqrt(2C)
// K comes score-expanded ({br,bi,-bi,br}): B_real=lo, B_imag=hi.
// Writes |S| (compact real) into areal[bh][n][m].
// =====================================================================
__global__ __launch_bounds__(32) void attn_scores(
    const float* __restrict__ q, const float* __restrict__ kbx,
    float* __restrict__ areal) {
  const int tn  = blockIdx.x;
  const int tm2 = blockIdx.y;
  const int bh  = blockIdx.z;
  const int b   = bh / HH, h = bh % HH;
  const int lane = threadIdx.x;
  const int half = lane >> 4, l16 = lane & 15;

  const int nrow = tn * 16 + l16;
  const int m0   = tm2 * 32 + l16;
  const int m1   = m0 + 16;

  const float* __restrict__ pa  = q + 2 * ((size_t)(b * NN + nrow) * EE + h * CC) + 2 * half;
  const float* __restrict__ pb0 = kbx + ((size_t)(b * NN + m0) * EE + h * CC + half) * 4;
  const float* __restrict__ pb1 = kbx + ((size_t)(b * NN + m1) * EE + h * CC + half) * 4;

  v8f aR0 = {}, aI0 = {}, aR1 = {}, aI1 = {};
  v2f aC[4]; v4f b0C[4], b1C[4];
#pragma unroll
  for (int u = 0; u < 4; ++u) {
    aC[u]  = *(const v2f*)(pa + 4 * u);
    b0C[u] = *(const v4f*)(pb0 + 8 * u);
    b1C[u] = *(const v4f*)(pb1 + 8 * u);
  }
  const int iters = CC / 8;               // 8
  for (int it = 0; it < iters - 1; ++it) {
    pa += 16; pb0 += 32; pb1 += 32;
    v2f aN[4]; v4f b0N[4], b1N[4];
#pragma unroll
    for (int u = 0; u < 4; ++u) {
      aN[u]  = *(const v2f*)(pa + 4 * u);
      b0N[u] = *(const v4f*)(pb0 + 8 * u);
      b1N[u] = *(const v4f*)(pb1 + 8 * u);
    }
#pragma unroll
    for (int u = 0; u < 4; ++u) {
      aR0 = wmma_f32(aC[u], vlo(b0C[u]), aR0);
      aI0 = wmma_f32(aC[u], vhi(b0C[u]), aI0);
      aR1 = wmma_f32(aC[u], vlo(b1C[u]), aR1);
      aI1 = wmma_f32(aC[u], vhi(b1C[u]), aI1);
    }
#pragma unroll
    for (int u = 0; u < 4; ++u) { aC[u] = aN[u]; b0C[u] = b0N[u]; b1C[u] = b1N[u]; }
  }
#pragma unroll
  for (int u = 0; u < 4; ++u) {
    aR0 = wmma_f32(aC[u], vlo(b0C[u]), aR0);
    aI0 = wmma_f32(aC[u], vhi(b0C[u]), aI0);
    aR1 = wmma_f32(aC[u], vlo(b1C[u]), aR1);
    aI1 = wmma_f32(aC[u], vhi(b1C[u]), aI1);
  }

  const size_t abase = (size_t)bh * NN * NN;
#pragma unroll
  for (int r = 0; r < 8; ++r) {
    const int n_ = tn * 16 + r + 8 * half;
    const float re0 = aR0[r] * INV_DIV, im0 = aI0[r] * INV_DIV;
    const float re1 = aR1[r] * INV_DIV, im1 = aI1[r] * INV_DIV;
    areal[abase + (size_t)n_ * NN + m0] = sqrtf(re0 * re0 + im0 * im0);
    areal[abase + (size_t)n_ * NN + m1] = sqrtf(re1 * re1 + im1 * im1);
  }
}

// =====================================================================
// Row softmax over |s|: reads compact areal, writes normalized prob back
// to areal (for the AV gemm) AND as dense complex {p,0} to the `a` output.
// =====================================================================
__global__ __launch_bounds__(256) void softmax_rows(
    float* __restrict__ areal, float* __restrict__ aout) {
  __shared__ float red[256];
  const size_t row = blockIdx.x;                 // 0 .. B*H*N-1
  float* __restrict__ pin = areal + row * (size_t)NN;
  const int t = threadIdx.x;

  float4 vv = *(const float4*)(pin + 4 * t);
  float vmax = fmaxf(fmaxf(vv.x, vv.y), fmaxf(vv.z, vv.w));
  red[t] = vmax;
  __syncthreads();
  for (int s = 128; s > 0; s >>= 1) {
    if (t < s) red[t] = fmaxf(red[t], red[t + s]);
    __syncthreads();
  }
  vmax = red[0];
  __syncthreads();

  vv.x = expf(vv.x - vmax); vv.y = expf(vv.y - vmax);
  vv.z = expf(vv.z - vmax); vv.w = expf(vv.w - vmax);
  red[t] = vv.x + vv.y + vv.z + vv.w;
  __syncthreads();
  for (int s = 128; s > 0; s >>= 1) {
    if (t < s) red[t] += red[t + s];
    __syncthreads();
  }
  const float inv = 1.0f / red[0];
  vv.x *= inv; vv.y *= inv; vv.z *= inv; vv.w *= inv;

  *(float4*)(pin + 4 * t) = vv;
  float* __restrict__ pc = aout + 2 * row * (size_t)NN + 8 * t;
  *(float4*)(pc)     = make_float4(vv.x, 0.0f, vv.y, 0.0f);
  *(float4*)(pc + 4) = make_float4(vv.z, 0.0f, vv.w, 0.0f);
}

// =====================================================================
// LDS-tiled transpose + de-interleave of v:
//   vtr[bh][c][m] = Re v[b][m][h*C+c],  vti[bh][c][m] = Im v[...]
// =====================================================================
__global__ void transpose_v(const float* __restrict__ vb,
                            float* __restrict__ vtr, float* __restrict__ vti) {
  __shared__ float tr[32][33];
  __shared__ float ti[32][33];
  const int bh = blockIdx.z, b = bh / HH, h = bh % HH;
  const int m0 = blockIdx.x * 32, c0 = blockIdx.y * 32;
  const int tx = threadIdx.x, ty = threadIdx.y;   // (32, 8)
#pragma unroll
  for (int i = 0; i < 4; ++i) {
    const int m = m0 + ty + 8 * i;
    const float2 vv = *(const float2*)(vb + 2 * ((size_t)(b * NN + m) * EE + h * CC + c0 + tx));
    tr[ty + 8 * i][tx] = vv.x;
    ti[ty + 8 * i][tx] = vv.y;
  }
  __syncthreads();
#pragma unroll
  for (int i = 0; i < 4; ++i) {
    const int c = c0 + ty + 8 * i;
    const size_t o = ((size_t)bh * CC + c) * NN + m0 + tx;
    vtr[o] = tr[tx][ty + 8 * i];
    vti[o] = ti[tx][ty + 8 * i];
  }
}

// =====================================================================
// av[n][c] = sum_m a[n][m] * conj(v[m][c]);  a real, v via transposed
// planes -> every fragment is a raw contiguous v2f load.
// One wave computes 16 rows x 32 c-columns; software-pipelined.
// =====================================================================
__global__ __launch_bounds__(32) void attn_av(
    const float* __restrict__ areal, const float* __restrict__ vtr,
    const float* __restrict__ vti, float* __restrict__ concat) {
  const int tn  = blockIdx.x;
  const int tc2 = blockIdx.y;
  const int bh  = blockIdx.z;
  const int b   = bh / HH, h = bh % HH;
  const int lane = threadIdx.x;
  const int half = lane >> 4, l16 = lane & 15;

  const int nrow = tn * 16 + l16;
  const int c0   = tc2 * 32 + l16;

  const float* __restrict__ pa  = areal + ((size_t)bh * NN + nrow) * NN + 2 * half;
  const float* __restrict__ pr0 = vtr + ((size_t)bh * CC + c0) * NN + 2 * half;
  const float* __restrict__ pi0 = vti + ((size_t)bh * CC + c0) * NN + 2 * half;
  const float* __restrict__ pr1 = pr0 + 16 * NN;
  const float* __restrict__ pi1 = pi0 + 16 * NN;

  v8f Pr0 = {}, Pi0 = {}, Pr1 = {}, Pi1 = {};
  v2f aC[4], r0C[4], i0C[4], r1C[4], i1C[4];
#pragma unroll
  for (int u = 0; u < 4; ++u) {
    aC[u]  = *(const v2f*)(pa + 4 * u);
    r0C[u] = *(const v2f*)(pr0 + 4 * u);
    i0C[u] = *(const v2f*)(pi0 + 4 * u);
    r1C[u] = *(const v2f*)(pr1 + 4 * u);
    i1C[u] = *(const v2f*)(pi1 + 4 * u);
  }
  const int iters = NN / 16;              // 64 (16 m-values per iteration)
  for (int it = 0; it < iters - 1; ++it) {
    pa += 16; pr0 += 16; pi0 += 16; pr1 += 16; pi1 += 16;
    v2f aN[4], r0N[4], i0N[4], r1N[4], i1N[4];
#pragma unroll
    for (int u = 0; u < 4; ++u) {
      aN[u]  = *(const v2f*)(pa + 4 * u);
      r0N[u] = *(const v2f*)(pr0 + 4 * u);
      i0N[u] = *(const v2f*)(pi0 + 4 * u);
      r1N[u] = *(const v2f*)(pr1 + 4 * u);
      i1N[u] = *(const v2f*)(pi1 + 4 * u);
    }
#pragma unroll
    for (int u = 0; u < 4; ++u) {
      Pr0 = wmma_f32(aC[u], r0C[u], Pr0);
      Pi0 = wmma_f32(aC[u], i0C[u], Pi0);
      Pr1 = wmma_f32(aC[u], r1C[u], Pr1);
      Pi1 = wmma_f32(aC[u], i1C[u], Pi1);
    }
#pragma unroll
    for (int u = 0; u < 4; ++u) {
      aC[u] = aN[u]; r0C[u] = r0N[u]; i0C[u] = i0N[u]; r1C[u] = r1N[u]; i1C[u] = i1N[u];
    }
  }
#pragma unroll
  for (int u = 0; u < 4; ++u) {
    Pr0 = wmma_f32(aC[u], r0C[u], Pr0);
    Pi0 = wmma_f32(aC[u], i0C[u], Pi0);
    Pr1 = wmma_f32(aC[u], r1C[u], Pr1);
    Pi1 = wmma_f32(aC[u], i1C[u], Pi1);
  }
#pragma unroll
  for (int r = 0; r < 8; ++r) {
    const int n_ = tn * 16 + r + 8 * half;
    float2* o0 = (float2*)(concat + 2 * ((size_t)(b * NN + n_) * EE + h * CC + c0));
    float2* o1 = (float2*)((float*)o0 + 32);
    *o0 = make_float2(Pr0[r], -Pi0[r]);           // conj(v)
    *o1 = make_float2(Pr1[r], -Pi1[r]);
  }
}

// =====================================================================
// Complex LayerNorm over F features with fused residual add.
// =====================================================================
__global__ __launch_bounds__(256) void complex_ln(
    const float* __restrict__ in1, const float* __restrict__ in2,
    const float* __restrict__ w, const float* __restrict__ bp,
    float* __restrict__ out) {
  __shared__ float redr[256];
  __shared__ float redi[256];
  const size_t row = blockIdx.x;
  const int t = threadIdx.x;
  const float2* __restrict__ p1 = (const float2*)(in1 + 2 * row * (size_t)FF);
  const float2* __restrict__ p2 = (const float2*)(in2 + 2 * row * (size_t)FF);

  float xr[2], xi[2], sr = 0.0f, si = 0.0f;
#pragma unroll
  for (int i = 0; i < 2; ++i) {
    const int idx = t + i * 256;
    const float2 a = p1[idx];
    const float2 c = p2[idx];
    xr[i] = a.x + c.x;
    xi[i] = a.y + c.y;
    sr += xr[i];
    si += xi[i];
  }
  redr[t] = sr; redi[t] = si;
  __syncthreads();
  for (int s = 128; s > 0; s >>= 1) {
    if (t < s) { redr[t] += redr[t + s]; redi[t] += redi[t + s]; }
    __syncthreads();
  }
  const float mr = redr[0] * (1.0f / FF);
  const float mi = redi[0] * (1.0f / FF);
  __syncthreads();

  float vs = 0.0f;
#pragma unroll
  for (int i = 0; i < 2; ++i) {
    xr[i] -= mr; xi[i] -= mi;
    vs += xr[i] * xr[i] + xi[i] * xi[i];
  }
  redr[t] = vs;
  __syncthreads();
  for (int s = 128; s > 0; s >>= 1) {
    if (t < s) redr[t] += redr[t + s];
    __syncthreads();
  }
  const float inv = rsqrtf(redr[0] * (1.0f / (FF - 1)));

  float2* __restrict__ po = (float2*)(out + 2 * row * (size_t)FF);
#pragma unroll
  for (int i = 0; i < 2; ++i) {
    const int idx = t + i * 256;
    const float nr = xr[i] * inv, ni = xi[i] * inv;
    const float2 wv = ((const float2*)w)[idx];
    const float2 bv = ((const float2*)bp)[idx];
    po[idx] = make_float2(wv.x * nr - wv.y * ni + bv.x,
                          wv.x * ni + wv.y * nr + bv.y);
  }
}

// =====================================================================
extern "C" void kernel_launch(void* const* d_in, const int* in_sizes, int n_in,
                              void* d_out, int out_size, void* d_ws, size_t ws_size,
                              hipStream_t stream) {
  (void)in_sizes; (void)n_in; (void)out_size; (void)ws_size;
  const float* x     = (const float*)d_in[0];
  const float* Wq    = (const float*)d_in[1];
  const float* bq    = (const float*)d_in[2];
  const float* Wk    = (const float*)d_in[3];
  const float* bk    = (const float*)d_in[4];
  const float* Wv    = (const float*)d_in[5];
  const float* bv    = (const float*)d_in[6];
  const float* Wcat  = (const float*)d_in[7];
  const float* bcat  = (const float*)d_in[8];
  const float* ln1w  = (const float*)d_in[9];
  const float* ln1b  = (const float*)d_in[10];
  const float* Wffn  = (const float*)d_in[11];
  const float* bffn  = (const float*)d_in[12];
  const float* ln2w  = (const float*)d_in[13];
  const float* ln2b  = (const float*)d_in[14];

  float* ws = (float*)d_ws;
  const size_t SZ  = (size_t)BN * EE * 2;         // (B,N,E) complex, floats
  const size_t WXZ = (size_t)EE * FF * 4;         // expanded weight, floats
  float* qb     = ws;                 ws += SZ;
  float* vb     = ws;                 ws += SZ;
  float* concat = ws;                 ws += SZ;
  float* tmp    = ws;                 ws += SZ;
  float* x1     = ws;                 ws += SZ;
  float* kbx    = ws;                 ws += (size_t)BN * EE * 4;   // expanded K
  float* vtr    = ws;                 ws += (size_t)BB * HH * CC * NN;
  float* vti    = ws;                 ws += (size_t)BB * HH * CC * NN;
  float* wqx    = ws;                 ws += WXZ;
  float* wkx    = ws;                 ws += WXZ;
  float* wvx    = ws;                 ws += WXZ;
  float* wcx    = ws;                 ws += WXZ;
  float* wfx    = ws;                 ws += WXZ;
  float* areal  = ws;                 ws += (size_t)BB * HH * NN * NN;

  float* x2out = (float*)d_out;                         // (B,N,F) complex
  float* aout  = x2out + 2 * (size_t)BB * NN * FF;      // (B,H,N,N) complex

  const dim3 blk32(32), blk256(256);
  const dim3 gExp((EE * FF) / 256);
  const dim3 gGemm((BN / 16) * (EE / 32));              // 4096 slab-waves

  // Expand weights once ({wr,-wi,wi,wr})
  expand_weight<<<gExp, blk256, 0, stream>>>(Wq,   wqx, EE * FF);
  expand_weight<<<gExp, blk256, 0, stream>>>(Wk,   wkx, EE * FF);
  expand_weight<<<gExp, blk256, 0, stream>>>(Wv,   wvx, EE * FF);
  expand_weight<<<gExp, blk256, 0, stream>>>(Wcat, wcx, FF * EE);
  expand_weight<<<gExp, blk256, 0, stream>>>(Wffn, wfx, FF * FF);

  // Q, K (score-expanded), V projections
  cgemm_nt_bias<<<gGemm, blk32, 0, stream>>>(x, wqx, bq, qb,  EE, FF, 0);
  cgemm_nt_bias<<<gGemm, blk32, 0, stream>>>(x, wkx, bk, kbx, EE, FF, 1);
  cgemm_nt_bias<<<gGemm, blk32, 0, stream>>>(x, wvx, bv, vb,  EE, FF, 0);

  // v -> transposed de-interleaved planes
  transpose_v<<<dim3(NN / 32, CC / 32, BB * HH), dim3(32, 8), 0, stream>>>(vb, vtr, vti);

  // scores -> |s| compact real
  attn_scores<<<dim3(NN / 16, NN / 32, BB * HH), blk32, 0, stream>>>(qb, kbx, areal);

  // softmax: areal (for AV) + dense complex a output
  softmax_rows<<<dim3(BB * HH * NN), blk256, 0, stream>>>(areal, aout);

  // av = a @ conj(v) -> concat layout
  attn_av<<<dim3(NN / 16, CC / 32, BB * HH), blk32, 0, stream>>>(areal, vtr, vti, concat);

  // attn_out = concat @ Wcat^T + bcat
  cgemm_nt_bias<<<gGemm, blk32, 0, stream>>>(concat, wcx, bcat, tmp, FF, EE, 0);

  // x1 = LN(attn_out + x)
  complex_ln<<<dim3(BN), blk256, 0, stream>>>(tmp, x, ln1w, ln1b, x1);

  // f = x1 @ Wffn^T + bffn
  cgemm_nt_bias<<<gGemm, blk32, 0, stream>>>(x1, wfx, bffn, tmp, FF, FF, 0);

  // x2 = LN(f + x1) -> first output
  complex_ln<<<dim3(BN), blk256, 0, stream>>>(tmp, x1, ln2w, ln2b, x2out);
}